// MSDAv3_46394236731951
// MI455X (gfx1250) — compile-verified
//
#include <hip/hip_runtime.h>
#include <hip/hip_bf16.h>
#include <math.h>

typedef __attribute__((ext_vector_type(2))) float v2f;
typedef __attribute__((ext_vector_type(8))) float v8f;
typedef __attribute__((ext_vector_type(4))) int   v4i;

#if defined(__has_builtin)
#if __has_builtin(__builtin_amdgcn_global_load_async_to_lds_b128) && \
    __has_builtin(__builtin_amdgcn_s_wait_asynccnt)
#define MSDA_USE_ASYNC_LDS 1
#endif
#endif
#ifndef MSDA_USE_ASYNC_LDS
#define MSDA_USE_ASYNC_LDS 0
#endif

// Builtin signature (from round-2 diagnostic): param0 = AS1 v4i*, param1 = AS3 v4i*
#define AS_G_V4I(p) ((__attribute__((address_space(1))) v4i*)(void*)(p))
#define AS_L_V4I(p) ((__attribute__((address_space(3))) v4i*)(void*)(p))

// ---------------------------------------------------------------------------
// fp32 GEMM with LDS-staged weights:  C[M,N] = A[M,K] @ W[N,K]^T + bias[N]
//
// Block = 256 threads = 8 waves. Block tile = 128 rows x 32 cols:
//  * Weight panel W[nb*32 .. nb*32+32, 0..K) (32 KB) is staged into LDS once,
//    via GLOBAL_LOAD_ASYNC_TO_LDS_B128 (ASYNCcnt) when available.
//  * Each wave owns one 16-row M tile and TWO adjacent 16-col N tiles:
//    one A fragment feeds two back-to-back V_WMMA_F32_16X16X4_F32 issues.
//  * B fragments come from LDS (ds_load_b64); A fragments stream from
//    global/L2 with global_prefetch_b8 lookahead.
// K must be 256 (true for all four GEMMs here).
// ---------------------------------------------------------------------------
__global__ void msda_gemm_lds(const float* __restrict__ A,
                              const float* __restrict__ W,
                              const float* __restrict__ bias,
                              float* __restrict__ C,
                              int M, int N, int K)
{
    __shared__ float Wlds[32 * 256];          // 32 KB weight panel

    const int tid = threadIdx.x;
    const int ntn = N >> 5;                   // # of 32-col panels
    const int nb  = blockIdx.x % ntn;
    const int mb  = blockIdx.x / ntn;

    // ---- stage weight panel: rows [nb*32, nb*32+32) x K -> LDS -------------
    const float* Wsrc = W + (size_t)(nb * 32) * K;     // 8192 floats = 32 KB
#if MSDA_USE_ASYNC_LDS
    // 256 threads x 16 B = 4 KB per issue; 8 issues cover 32 KB.
    #pragma unroll
    for (int i = 0; i < 8; ++i) {
        const int byteOff = (i * 256 + tid) * 16;
        __builtin_amdgcn_global_load_async_to_lds_b128(
            AS_G_V4I((const char*)Wsrc + byteOff),
            AS_L_V4I((char*)Wlds + byteOff),
            /*offset=*/0, /*cpol=*/0);
    }
    __builtin_amdgcn_s_wait_asynccnt(0);
    __syncthreads();
#else
    #pragma unroll
    for (int i = tid; i < (32 * 256) / 4; i += 256)
        ((float4*)Wlds)[i] = ((const float4*)Wsrc)[i];
    __syncthreads();
#endif

    // ---- per-wave 16x32 output tile ----------------------------------------
    const int lane = tid & 31;
    const int wv   = tid >> 5;
    const int tm   = mb * 8 + wv;
    if (tm * 16 >= M) return;                 // no barriers after this point

    const int r  = lane & 15;                 // row (A) / col (B) within tile
    const int kk = (lane >> 4) << 1;          // 0 or 2

    const float* __restrict__ Arow = A    + (size_t)(tm * 16 + r) * K + kk;
    const float* __restrict__ B0   = Wlds + (size_t)r        * K + kk;   // cols nb*32 + r
    const float* __restrict__ B1   = Wlds + (size_t)(16 + r) * K + kk;   // cols nb*32+16 + r

    v8f acc0 = {};
    v8f acc1 = {};
    #pragma unroll 4
    for (int k = 0; k < K; k += 4) {
        if ((k & 63) == 0 && k + 64 < K)
            __builtin_prefetch(Arow + k + 64, 0, 0);
        v2f a, b0, b1;
        a.x  = Arow[k]; a.y  = Arow[k + 1];
        b0.x = B0[k];   b0.y = B0[k + 1];
        b1.x = B1[k];   b1.y = B1[k + 1];
        acc0 = __builtin_amdgcn_wmma_f32_16x16x4_f32(
                   false, a, false, b0, (short)0, acc0, false, false);
        acc1 = __builtin_amdgcn_wmma_f32_16x16x4_f32(
                   false, a, false, b1, (short)0, acc1, false, false);
    }

    const int   col0 = nb * 32 + r;
    const int   row0 = tm * 16 + ((lane >> 4) << 3);
    const float bb0  = bias[col0];
    const float bb1  = bias[col0 + 16];
    #pragma unroll
    for (int v = 0; v < 8; ++v) {
        float* crow = C + (size_t)(row0 + v) * N + col0;
        crow[0]  = acc0[v] + bb0;
        crow[16] = acc1[v] + bb1;
    }
}

// ---------------------------------------------------------------------------
// Bilinear sampling + 20-point softmax attention.
// One wave32 per (b, n, h). lane = channel: key ch = lane, val ch = 32+lane.
// kv stays in natural [B*S, 512] layout (head transpose is pure indexing);
// the whole kv tensor (89 MB) is L2-resident (192 MB), so gathers hit L2.
// ---------------------------------------------------------------------------
__global__ void msda_sample_attn(const float* __restrict__ priors,  // [B,N,L,2]
                                 const float* __restrict__ off,     // [B*N,320]
                                 const float* __restrict__ qbuf,    // [B*N,256]
                                 const float* __restrict__ kv,      // [B*S,512]
                                 const float* __restrict__ pe,      // [8,20,32]
                                 const int*   __restrict__ shapes,  // [5,2]
                                 const int*   __restrict__ starts,  // [5]
                                 float*       __restrict__ outw,    // [B*N,256]
                                 int B, int N, int S)
{
    const int Hh = 8, L = 5, P = 4, NP = 20;
    const int lane = threadIdx.x & 31;
    const int unit = blockIdx.x * (blockDim.x >> 5) + (threadIdx.x >> 5);
    if (unit >= B * N * Hh) return;

    const int h  = unit & (Hh - 1);
    const int bn = unit >> 3;
    const int b  = bn / N;

    const float qs = qbuf[(size_t)bn * 256 + h * 32 + lane] * 0.17677669529663687f;

    float logit[NP];
    float vals [NP];

    #pragma unroll
    for (int p = 0; p < NP; ++p) {
        const int l  = p >> 2;
        const int pp = p & 3;
        // shapes_wh = shapes[:, ::-1]  ->  W = shapes[l][1], H = shapes[l][0]
        const int   Wd = shapes[l * 2 + 1];
        const int   Hd = shapes[l * 2 + 0];
        const float Wf = (float)Wd, Hf = (float)Hd;

        const float px = priors[((size_t)bn * L + l) * 2 + 0];
        const float py = priors[((size_t)bn * L + l) * 2 + 1];
        const int   oi = ((h * L + l) * P + pp) * 2;
        const float ox = off[(size_t)bn * 320 + oi + 0];
        const float oy = off[(size_t)bn * 320 + oi + 1];

        const float x  = px * Wf + ox - 0.5f;   // (prior + off/W)*W - 0.5
        const float y  = py * Hf + oy - 0.5f;
        const float x0 = floorf(x);
        const float y0 = floorf(y);

        const int st = starts[l];
        float key = 0.f, val = 0.f;
        #pragma unroll
        for (int dy = 0; dy < 2; ++dy) {
            #pragma unroll
            for (int dx = 0; dx < 2; ++dx) {
                const float xi = x0 + (float)dx;
                const float yi = y0 + (float)dy;
                const bool valid = (xi >= 0.f) && (xi < Wf) && (yi >= 0.f) && (yi < Hf);
                if (valid) {
                    const float wgt = (1.f - fabsf(x - xi)) * (1.f - fabsf(y - yi));
                    const int xic = (int)xi;
                    const int yic = (int)yi;
                    const size_t base =
                        ((size_t)(b * S + st + yic * Wd + xic)) * 512 + (size_t)h * 64;
                    key += wgt * kv[base + lane];
                    val += wgt * kv[base + 32 + lane];
                }
            }
        }
        key += pe[((size_t)h * NP + p) * 32 + lane];

        // wave32 dot-product reduction (ds_swizzle / permute path)
        float part = key * qs;
        #pragma unroll
        for (int o = 16; o > 0; o >>= 1)
            part += __shfl_xor(part, o, 32);
        logit[p] = part;
        vals[p]  = val;
    }

    float m = logit[0];
    #pragma unroll
    for (int p = 1; p < NP; ++p) m = fmaxf(m, logit[p]);
    float sum = 0.f;
    #pragma unroll
    for (int p = 0; p < NP; ++p) { logit[p] = __expf(logit[p] - m); sum += logit[p]; }
    const float inv = 1.f / sum;

    float o = 0.f;
    #pragma unroll
    for (int p = 0; p < NP; ++p) o += logit[p] * vals[p];

    outw[(size_t)bn * 256 + h * 32 + lane] = o * inv;
}

// ---------------------------------------------------------------------------
// Host-side orchestration
// ---------------------------------------------------------------------------
extern "C" void kernel_launch(void* const* d_in, const int* in_sizes, int n_in,
                              void* d_out, int out_size, void* d_ws, size_t ws_size,
                              hipStream_t stream)
{
    (void)in_sizes; (void)n_in; (void)out_size; (void)ws_size;

    const float* in_feats     = (const float*)d_in[0];   // [B,N,256]
    const float* priors       = (const float*)d_in[1];   // [B,N,5,2]
    const float* sample_feats = (const float*)d_in[2];   // [B,S,256]
    const int*   shapes       = (const int*)  d_in[3];   // [5,2]
    const int*   starts       = (const int*)  d_in[4];   // [5]
    const float* W_off        = (const float*)d_in[5];   // [320,256]
    const float* b_off        = (const float*)d_in[6];   // [320]
    const float* W_q          = (const float*)d_in[7];   // [256,256]
    const float* b_q          = (const float*)d_in[8];   // [256]
    const float* W_kv         = (const float*)d_in[9];   // [512,256]
    const float* b_kv         = (const float*)d_in[10];  // [512]
    const float* pe           = (const float*)d_in[11];  // [8,20,32]
    const float* W_out        = (const float*)d_in[12];  // [256,256]
    const float* b_out        = (const float*)d_in[13];  // [256]

    const int B = 2, N = 2048, S = 21824, K = 256;
    const int BN = B * N;                 // 4096

    // Workspace layout (fp32): kv | off | q | weighted
    float* kv_ws  = (float*)d_ws;                         // [B*S, 512]  89.4 MB
    float* off_ws = kv_ws  + (size_t)B * S * 512;         // [BN, 320]    5.2 MB
    float* q_ws   = off_ws + (size_t)BN * 320;            // [BN, 256]    4.2 MB
    float* w_ws   = q_ws   + (size_t)BN * 256;            // [BN, 256]    4.2 MB

    auto gemm = [&](const float* A, const float* W, const float* bias,
                    float* C, int M, int Nout) {
        const int ntn = Nout / 32;                        // 32-col panels
        const int mbs = (M / 16 + 7) / 8;                 // 8 M-tiles per block
        msda_gemm_lds<<<dim3(mbs * ntn), 256, 0, stream>>>(A, W, bias, C, M, Nout, K);
    };

    gemm(in_feats,     W_off, b_off, off_ws, BN,    320);   // offsets
    gemm(in_feats,     W_q,   b_q,   q_ws,   BN,    256);   // queries
    gemm(sample_feats, W_kv,  b_kv,  kv_ws,  B * S, 512);   // keys+values

    const int units = BN * 8;                               // 32768 (b,n,h) waves
    msda_sample_attn<<<units / 8, 256, 0, stream>>>(
        priors, off_ws, q_ws, kv_ws, pe, shapes, starts, w_ws, B, N, S);

    gemm(w_ws, W_out, b_out, (float*)d_out, BN, 256);       // output projection
}